// DirectedEdgeEncoder_43722767073861
// MI455X (gfx1250) — compile-verified
//
#include <hip/hip_runtime.h>

typedef __attribute__((ext_vector_type(2))) float v2f;
typedef __attribute__((ext_vector_type(8))) float v8f;

#define IN_F   128
#define HID_F  64
#define OUT_F  32

// ---------------- degree + norm precompute ----------------

__global__ void k_init_deg(float* __restrict__ deg_in, float* __restrict__ deg_out, int N) {
  int i = blockIdx.x * blockDim.x + threadIdx.x;
  if (i < N) { deg_in[i] = 1.0f; deg_out[i] = 1.0f; }  // +1 self loop
}

__global__ void k_count_deg(const int* __restrict__ ei0, const int* __restrict__ ei1,
                            float* __restrict__ deg_in, float* __restrict__ deg_out, int E) {
  int e = blockIdx.x * blockDim.x + threadIdx.x;
  if (e < E) {
    atomicAdd(&deg_out[ei0[e]], 1.0f);
    atomicAdd(&deg_in [ei1[e]], 1.0f);
  }
}

// rn_edge[e] = (deg_in[dst]*deg_out[src])^-1/2  (same value fwd & bwd since alpha=beta=0.5)
__global__ void k_norms(const int* __restrict__ ei0, const int* __restrict__ ei1,
                        const float* __restrict__ deg_in, const float* __restrict__ deg_out,
                        float* __restrict__ rn_edge, float* __restrict__ rn_self, int E, int N) {
  int i = blockIdx.x * blockDim.x + threadIdx.x;
  if (i < E) {
    rn_edge[i] = rsqrtf(deg_in[ei1[i]] * deg_out[ei0[i]]);
  } else if (i < E + N) {
    int n = i - E;
    rn_self[n] = rsqrtf(deg_in[n] * deg_out[n]);
  }
}

// ---------------- WMMA fp32 GEMM ----------------
// Y[n][c] = sum_k X[n][koff+k] * W[c'][k] + bias[c'], two weight blocks (a/b),
// second block reads X at KOFFB. One wave = one 16-row M tile across ALL column
// tiles; the per-lane A row is preloaded into registers once (K compile-time so
// afrag[] stays in VGPRs and the K/4-deep WMMA chain fully unrolls).

template <int K, int RELU>
__global__ __launch_bounds__(128)
void k_gemm_wmma(const float* __restrict__ X, int ldx,
                 const float* __restrict__ Wa, const float* __restrict__ ba,
                 const float* __restrict__ Wb, const float* __restrict__ bb,
                 float* __restrict__ Y, int ldy,
                 int halfO, int koffB, int nColTiles, int nMTiles) {
  constexpr int NK = K / 4;

  int wave = threadIdx.x >> 5;
  int lane = threadIdx.x & 31;
  int mt   = blockIdx.x * 4 + wave;
  if (mt >= nMTiles) return;               // wave-uniform: EXEC all-1s inside WMMA

  int m0   = mt << 4;
  int mrow = lane & 15;
  int khi  = (lane >> 4) << 1;             // lanes 0-15: K 0,1 ; lanes 16-31: K 2,3
  int col  = lane & 15;

  const float* xrow = X + (size_t)(m0 + mrow) * ldx;

  v2f afrag[NK];
  int curKoff = -1;

  for (int ct = 0; ct < nColTiles; ++ct) {
    int c0   = ct << 4;
    int half = (c0 >= halfO) ? 1 : 0;
    const float* W    = half ? Wb : Wa;
    const float* bias = half ? bb : ba;
    int wc0  = c0 - half * halfO;
    int koff = half ? koffB : 0;

    if (koff != curKoff) {                 // A preload: once for layer1, twice for layer2
      curKoff = koff;
      const float* xp = xrow + koff + khi;
#pragma unroll
      for (int i = 0; i < NK; ++i) {
        v2f a = *(const v2f*)(xp + i * 4);
        if (RELU) { a.x = fmaxf(a.x, 0.0f); a.y = fmaxf(a.y, 0.0f); }
        afrag[i] = a;
      }
    }

    const float* wrow = W + (size_t)(wc0 + col) * K + khi;
    v8f acc = {};
#pragma unroll
    for (int i = 0; i < NK; ++i) {
      v2f bf = *(const v2f*)(wrow + i * 4);
      acc = __builtin_amdgcn_wmma_f32_16x16x4_f32(false, afrag[i], false, bf,
                                                  (short)0, acc, false, false);
    }

    float bv = bias[wc0 + col];
    int mbase = m0 + ((lane >> 4) << 3);   // C/D: vgpr g -> row g (lanes<16) / g+8 (lanes>=16)
    float* yp = Y + (size_t)mbase * ldy + c0 + col;
#pragma unroll
    for (int g = 0; g < 8; ++g)
      yp[(size_t)g * ldy] = acc[g] + bv;
  }
}

// ---------------- aggregation: Y = rn_self .* T  then  Y[dst] += rn_edge * T[src] -------------

__global__ void k_aggr_self(const float* __restrict__ T, const float* __restrict__ rn_self,
                            float* __restrict__ Y, int total4, int fshift) {
  int idx = blockIdx.x * blockDim.x + threadIdx.x;   // float4 granularity; F>=64 so one node
  if (idx < total4) {
    float rn = rn_self[(idx << 2) >> fshift];
    float4 v = ((const float4*)T)[idx];
    float4 o; o.x = rn * v.x; o.y = rn * v.y; o.z = rn * v.z; o.w = rn * v.w;
    ((float4*)Y)[idx] = o;
  }
}

__global__ void k_scatter(const float* __restrict__ T,
                          const int* __restrict__ src, const int* __restrict__ dst,
                          const float* __restrict__ rn_edge, float* __restrict__ Y,
                          int E, int F) {
  int chunks = F >> 2;                     // float4 per lane; wave32 covers 1 node row @F=128
  int tid = blockIdx.x * blockDim.x + threadIdx.x;
  if (tid >= E * chunks) return;
  int e = tid / chunks;
  int c = (tid - e * chunks) << 2;
  float rn = rn_edge[e];
  const float4 v = *(const float4*)(T + (size_t)src[e] * F + c);
  float* yp = Y + (size_t)dst[e] * F + c;
  atomicAdd(yp + 0, rn * v.x);
  atomicAdd(yp + 1, rn * v.y);
  atomicAdd(yp + 2, rn * v.z);
  atomicAdd(yp + 3, rn * v.w);
}

// ---------------- reparametrize: out = mu + noise * exp(logstd) / 5 ----------------

__global__ void k_final(const float* __restrict__ A2s, const float* __restrict__ A2t,
                        const float* __restrict__ noise_s, const float* __restrict__ noise_t,
                        float* __restrict__ out, int N) {
  int idx = blockIdx.x * blockDim.x + threadIdx.x;
  int total = N * OUT_F;
  if (idx < total) {
    int n = idx >> 5, j = idx & 31;
    out[idx] = A2s[n * 64 + j] + noise_s[idx] * expf(A2s[n * 64 + 32 + j]) * 0.2f;
  } else if (idx < 2 * total) {
    int k = idx - total;
    int n = k >> 5, j = k & 31;
    out[idx] = A2t[n * 64 + j] + noise_t[k] * expf(A2t[n * 64 + 32 + j]) * 0.2f;
  }
}

// ---------------- driver ----------------

static inline int ceil_div(int a, int b) { return (a + b - 1) / b; }

extern "C" void kernel_launch(void* const* d_in, const int* in_sizes, int n_in,
                              void* d_out, int out_size, void* d_ws, size_t ws_size,
                              hipStream_t stream) {
  const float* s       = (const float*)d_in[0];
  const float* t       = (const float*)d_in[1];
  const int*   ei      = (const int*)  d_in[2];
  const float* noise_s = (const float*)d_in[3];
  const float* noise_t = (const float*)d_in[4];
  const float* sm1_W = (const float*)d_in[5],  *sm1_b = (const float*)d_in[6];
  const float* sm2_W = (const float*)d_in[7],  *sm2_b = (const float*)d_in[8];
  const float* sl1_W = (const float*)d_in[9],  *sl1_b = (const float*)d_in[10];
  const float* sl2_W = (const float*)d_in[11], *sl2_b = (const float*)d_in[12];
  const float* tm1_W = (const float*)d_in[13], *tm1_b = (const float*)d_in[14];
  const float* tm2_W = (const float*)d_in[15], *tm2_b = (const float*)d_in[16];
  const float* tl1_W = (const float*)d_in[17], *tl1_b = (const float*)d_in[18];
  const float* tl2_W = (const float*)d_in[19], *tl2_b = (const float*)d_in[20];

  const int N = in_sizes[0] / IN_F;
  const int E = in_sizes[2] / 2;
  const int* ei0 = ei;       // sources
  const int* ei1 = ei + E;   // targets

  float* ws      = (float*)d_ws;
  float* deg_in  = ws;
  float* deg_out = deg_in  + N;
  float* rn_self = deg_out + N;
  float* rn_edge = rn_self + N;
  float* Y1s = rn_edge + E;                   // [N,128] lin1 (sm|sl)
  float* Y1t = Y1s + (size_t)N * IN_F;        // [N,128] lin1 (tm|tl)
  float* A1s = Y1t + (size_t)N * IN_F;        // [N,128] aggregated
  float* A1t = A1s + (size_t)N * IN_F;
  float* Y2s = Y1s;                           // reuse: [N,64] lin2 (sm2|sl2)
  float* A2s = Y1s + (size_t)N * HID_F;       // [N,64] mu|logstd (s)
  float* Y2t = Y1t;
  float* A2t = Y1t + (size_t)N * HID_F;

  const int B = 256;
  const int nMTiles = ceil_div(N, 16);
  const int gemmBlocks = ceil_div(nMTiles, 4);

  // degrees + norms (shared by all 8 conv applications)
  k_init_deg<<<ceil_div(N, B), B, 0, stream>>>(deg_in, deg_out, N);
  k_count_deg<<<ceil_div(E, B), B, 0, stream>>>(ei0, ei1, deg_in, deg_out, E);
  k_norms<<<ceil_div(E + N, B), B, 0, stream>>>(ei0, ei1, deg_in, deg_out, rn_edge, rn_self, E, N);

  // layer 1 GEMMs (fp32 WMMA): [N,128] x [128,128] (two 64-wide weight blocks)
  k_gemm_wmma<IN_F, 0><<<gemmBlocks, 128, 0, stream>>>(
      s, IN_F, sm1_W, sm1_b, sl1_W, sl1_b, Y1s, IN_F, HID_F, 0, IN_F / 16, nMTiles);
  k_gemm_wmma<IN_F, 0><<<gemmBlocks, 128, 0, stream>>>(
      t, IN_F, tm1_W, tm1_b, tl1_W, tl1_b, Y1t, IN_F, HID_F, 0, IN_F / 16, nMTiles);

  // layer 1 aggregation: s-side forward (src=ei0,dst=ei1), t-side backward
  {
    int total4 = N * IN_F / 4, work = E * (IN_F >> 2);
    k_aggr_self<<<ceil_div(total4, B), B, 0, stream>>>(Y1s, rn_self, A1s, total4, 7);
    k_scatter  <<<ceil_div(work,   B), B, 0, stream>>>(Y1s, ei0, ei1, rn_edge, A1s, E, IN_F);
    k_aggr_self<<<ceil_div(total4, B), B, 0, stream>>>(Y1t, rn_self, A1t, total4, 7);
    k_scatter  <<<ceil_div(work,   B), B, 0, stream>>>(Y1t, ei1, ei0, rn_edge, A1t, E, IN_F);
  }

  // layer 2 GEMMs with ReLU-on-load: [N,64]x[64,32] per encoder, packed as dual blocks
  k_gemm_wmma<HID_F, 1><<<gemmBlocks, 128, 0, stream>>>(
      A1s, IN_F, sm2_W, sm2_b, sl2_W, sl2_b, Y2s, HID_F, OUT_F, HID_F, HID_F / 16, nMTiles);
  k_gemm_wmma<HID_F, 1><<<gemmBlocks, 128, 0, stream>>>(
      A1t, IN_F, tm2_W, tm2_b, tl2_W, tl2_b, Y2t, HID_F, OUT_F, HID_F, HID_F / 16, nMTiles);

  // layer 2 aggregation: s-side backward, t-side forward
  {
    int total4 = N * HID_F / 4, work = E * (HID_F >> 2);
    k_aggr_self<<<ceil_div(total4, B), B, 0, stream>>>(Y2s, rn_self, A2s, total4, 6);
    k_scatter  <<<ceil_div(work,   B), B, 0, stream>>>(Y2s, ei1, ei0, rn_edge, A2s, E, HID_F);
    k_aggr_self<<<ceil_div(total4, B), B, 0, stream>>>(Y2t, rn_self, A2t, total4, 6);
    k_scatter  <<<ceil_div(work,   B), B, 0, stream>>>(Y2t, ei0, ei1, rn_edge, A2t, E, HID_F);
  }

  // reparametrize into d_out: [s_out | t_out]
  k_final<<<ceil_div(2 * N * OUT_F, B), B, 0, stream>>>(A2s, A2t, noise_s, noise_t,
                                                        (float*)d_out, N);
}